// CLAM_42674795053600
// MI455X (gfx1250) — compile-verified
//
#include <hip/hip_runtime.h>
#include <hip/hip_bf16.h>
#include <math.h>

#define Nn 131072
#define Ee 1024
#define Ll 512
#define Dd 256
#define NI 50

typedef float v2f __attribute__((ext_vector_type(2)));
typedef float v8f __attribute__((ext_vector_type(8)));

__device__ __forceinline__ v8f wmma4(v2f a, v2f b, v8f c) {
  // D = A(16x4 f32) * B(4x16 f32) + C(16x16 f32)
  return __builtin_amdgcn_wmma_f32_16x16x4_f32(false, a, false, b, (short)0, c,
                                               false, false);
}
__device__ __forceinline__ v2f ld2(const float* p) { return *(const v2f*)p; }

__device__ __forceinline__ unsigned ordf(float f) {
  unsigned u = __float_as_uint(f);
  return (u & 0x80000000u) ? ~u : (u | 0x80000000u);
}

// ---------------- Kernel 1: h = relu(x @ Wc^T + bc)  [N,1024]x[1024,512]
__global__ __launch_bounds__(256) void k_gemm_h(const float* __restrict__ x,
                                                const float* __restrict__ Wc,
                                                const float* __restrict__ bc,
                                                float* __restrict__ h) {
  const int lane = threadIdx.x & 31;
  const int wave = threadIdx.x >> 5;
  const int wm = wave >> 1;  // 0..3
  const int wn = wave & 1;   // 0..1
  const int m15 = lane & 15;
  const int khalf = (lane >> 4) << 1;  // 0 or 2
  const int rowBase = blockIdx.y * 128 + wm * 32;
  const int colBase = blockIdx.x * 64 + wn * 32;

  v8f acc[2][2];
#pragma unroll
  for (int m = 0; m < 2; ++m)
#pragma unroll
    for (int n = 0; n < 2; ++n) acc[m][n] = v8f{0.f, 0.f, 0.f, 0.f, 0.f, 0.f, 0.f, 0.f};

  const float* xp = x + (size_t)(rowBase + m15) * Ee + khalf;
  const float* wp = Wc + (size_t)(colBase + m15) * Ee + khalf;

#pragma unroll 4
  for (int k = 0; k < Ee; k += 4) {
    v2f a0 = ld2(xp + k);
    v2f a1 = ld2(xp + (size_t)16 * Ee + k);
    v2f b0 = ld2(wp + k);
    v2f b1 = ld2(wp + (size_t)16 * Ee + k);
    acc[0][0] = wmma4(a0, b0, acc[0][0]);
    acc[0][1] = wmma4(a0, b1, acc[0][1]);
    acc[1][0] = wmma4(a1, b0, acc[1][0]);
    acc[1][1] = wmma4(a1, b1, acc[1][1]);
  }

#pragma unroll
  for (int m = 0; m < 2; ++m)
#pragma unroll
    for (int n = 0; n < 2; ++n) {
      const int col = colBase + n * 16 + m15;
      const float bcv = bc[col];
      const int row0 = rowBase + m * 16 + ((lane >> 4) << 3);
#pragma unroll
      for (int r = 0; r < 8; ++r) {
        float v = acc[m][n][r] + bcv;
        h[(size_t)(row0 + r) * Ll + col] = fmaxf(v, 0.f);
      }
    }
}

// ---------------- Kernel 2: g = tanh(h@Wv^T+bv)*sigmoid(h@Wu^T+bu)
__global__ __launch_bounds__(256) void k_gemm_g(const float* __restrict__ h,
                                                const float* __restrict__ Wv,
                                                const float* __restrict__ bv,
                                                const float* __restrict__ Wu,
                                                const float* __restrict__ bu,
                                                float* __restrict__ g) {
  const int lane = threadIdx.x & 31;
  const int wave = threadIdx.x >> 5;
  const int wm = wave >> 1;
  const int wn = wave & 1;
  const int m15 = lane & 15;
  const int khalf = (lane >> 4) << 1;
  const int rowBase = blockIdx.y * 128 + wm * 32;
  const int colBase = blockIdx.x * 64 + wn * 32;  // over D=256

  v8f aV[2][2], aU[2][2];
#pragma unroll
  for (int m = 0; m < 2; ++m)
#pragma unroll
    for (int n = 0; n < 2; ++n) {
      aV[m][n] = v8f{0.f, 0.f, 0.f, 0.f, 0.f, 0.f, 0.f, 0.f};
      aU[m][n] = v8f{0.f, 0.f, 0.f, 0.f, 0.f, 0.f, 0.f, 0.f};
    }

  const float* hp = h + (size_t)(rowBase + m15) * Ll + khalf;
  const float* vp = Wv + (size_t)(colBase + m15) * Ll + khalf;
  const float* up = Wu + (size_t)(colBase + m15) * Ll + khalf;

#pragma unroll 2
  for (int k = 0; k < Ll; k += 4) {
    v2f a0 = ld2(hp + k);
    v2f a1 = ld2(hp + (size_t)16 * Ll + k);
    v2f v0 = ld2(vp + k);
    v2f v1 = ld2(vp + (size_t)16 * Ll + k);
    v2f u0 = ld2(up + k);
    v2f u1 = ld2(up + (size_t)16 * Ll + k);
    aV[0][0] = wmma4(a0, v0, aV[0][0]);
    aV[0][1] = wmma4(a0, v1, aV[0][1]);
    aV[1][0] = wmma4(a1, v0, aV[1][0]);
    aV[1][1] = wmma4(a1, v1, aV[1][1]);
    aU[0][0] = wmma4(a0, u0, aU[0][0]);
    aU[0][1] = wmma4(a0, u1, aU[0][1]);
    aU[1][0] = wmma4(a1, u0, aU[1][0]);
    aU[1][1] = wmma4(a1, u1, aU[1][1]);
  }

#pragma unroll
  for (int m = 0; m < 2; ++m)
#pragma unroll
    for (int n = 0; n < 2; ++n) {
      const int col = colBase + n * 16 + m15;
      const float bvv = bv[col], buv = bu[col];
      const int row0 = rowBase + m * 16 + ((lane >> 4) << 3);
#pragma unroll
      for (int r = 0; r < 8; ++r) {
        float t = tanhf(aV[m][n][r] + bvv);
        float su = aU[m][n][r] + buv;
        float s = 1.f / (1.f + expf(-su));
        g[(size_t)(row0 + r) * Dd + col] = t * s;
      }
    }
}

// ---------------- Kernel 3: A = g@Wa^T+ba ; raw_attention out ; selection copies
__global__ __launch_bounds__(256) void k_proj(const float* __restrict__ g,
                                              const float* __restrict__ Wa,
                                              const float* __restrict__ ba,
                                              const int* __restrict__ plabel,
                                              float* __restrict__ A,
                                              float* __restrict__ vt,
                                              float* __restrict__ vb,
                                              float* __restrict__ out_raw) {
  __shared__ float wa[2 * Dd];
  for (int j = threadIdx.x; j < 2 * Dd; j += 256) wa[j] = Wa[j];
  __syncthreads();
  const int i = blockIdx.x * 256 + threadIdx.x;
  const float4* gr = (const float4*)(g + (size_t)i * Dd);
  const float4* w0 = (const float4*)(wa);
  const float4* w1 = (const float4*)(wa + Dd);
  float a0 = ba[0], a1 = ba[1];
#pragma unroll 4
  for (int k = 0; k < Dd / 4; ++k) {
    float4 v = gr[k];
    float4 x0 = w0[k], x1 = w1[k];
    a0 += v.x * x0.x + v.y * x0.y + v.z * x0.z + v.w * x0.w;
    a1 += v.x * x1.x + v.y * x1.y + v.z * x1.z + v.w * x1.w;
  }
  A[2 * i] = a0;
  A[2 * i + 1] = a1;
  out_raw[i] = a0;
  out_raw[Nn + i] = a1;
  const int lab = *plabel;
  const float av = (lab == 0) ? a0 : a1;
  vt[i] = av;
  vb[i] = av;
}

// ---------------- softmax normalizers (deterministic two-stage)
__global__ __launch_bounds__(256) void k_max1(const float* __restrict__ A,
                                              float* __restrict__ pmax) {
  float m0 = -INFINITY, m1 = -INFINITY;
  for (int i = blockIdx.x * 256 + threadIdx.x; i < Nn; i += 256 * 256) {
    m0 = fmaxf(m0, A[2 * i]);
    m1 = fmaxf(m1, A[2 * i + 1]);
  }
  __shared__ float s0[256], s1[256];
  s0[threadIdx.x] = m0;
  s1[threadIdx.x] = m1;
  __syncthreads();
  for (int s = 128; s > 0; s >>= 1) {
    if (threadIdx.x < s) {
      s0[threadIdx.x] = fmaxf(s0[threadIdx.x], s0[threadIdx.x + s]);
      s1[threadIdx.x] = fmaxf(s1[threadIdx.x], s1[threadIdx.x + s]);
    }
    __syncthreads();
  }
  if (threadIdx.x == 0) {
    pmax[blockIdx.x * 2] = s0[0];
    pmax[blockIdx.x * 2 + 1] = s1[0];
  }
}

__global__ __launch_bounds__(256) void k_max2(const float* __restrict__ pmax,
                                              float* __restrict__ mx) {
  __shared__ float s0[256], s1[256];
  s0[threadIdx.x] = pmax[2 * threadIdx.x];
  s1[threadIdx.x] = pmax[2 * threadIdx.x + 1];
  __syncthreads();
  for (int s = 128; s > 0; s >>= 1) {
    if (threadIdx.x < s) {
      s0[threadIdx.x] = fmaxf(s0[threadIdx.x], s0[threadIdx.x + s]);
      s1[threadIdx.x] = fmaxf(s1[threadIdx.x], s1[threadIdx.x + s]);
    }
    __syncthreads();
  }
  if (threadIdx.x == 0) {
    mx[0] = s0[0];
    mx[1] = s1[0];
  }
}

__global__ __launch_bounds__(256) void k_sum1(const float* __restrict__ A,
                                              const float* __restrict__ mx,
                                              float* __restrict__ psum) {
  const float m0 = mx[0], m1 = mx[1];
  float a0 = 0.f, a1 = 0.f;
  for (int i = blockIdx.x * 256 + threadIdx.x; i < Nn; i += 256 * 256) {
    a0 += expf(A[2 * i] - m0);
    a1 += expf(A[2 * i + 1] - m1);
  }
  __shared__ float s0[256], s1[256];
  s0[threadIdx.x] = a0;
  s1[threadIdx.x] = a1;
  __syncthreads();
  for (int s = 128; s > 0; s >>= 1) {
    if (threadIdx.x < s) {
      s0[threadIdx.x] += s0[threadIdx.x + s];
      s1[threadIdx.x] += s1[threadIdx.x + s];
    }
    __syncthreads();
  }
  if (threadIdx.x == 0) {
    psum[blockIdx.x * 2] = s0[0];
    psum[blockIdx.x * 2 + 1] = s1[0];
  }
}

__global__ __launch_bounds__(256) void k_sum2(const float* __restrict__ psum,
                                              float* __restrict__ sm) {
  __shared__ float s0[256], s1[256];
  s0[threadIdx.x] = psum[2 * threadIdx.x];
  s1[threadIdx.x] = psum[2 * threadIdx.x + 1];
  __syncthreads();
  for (int s = 128; s > 0; s >>= 1) {
    if (threadIdx.x < s) {
      s0[threadIdx.x] += s0[threadIdx.x + s];
      s1[threadIdx.x] += s1[threadIdx.x + s];
    }
    __syncthreads();
  }
  if (threadIdx.x == 0) {
    sm[0] = s0[0];
    sm[1] = s1[0];
  }
}

// ---------------- top-50 / bottom-50 (single block, 100 argmax passes)
__global__ __launch_bounds__(1024) void k_topk(float* __restrict__ vt,
                                               float* __restrict__ vb,
                                               int* __restrict__ idx) {
  const int tid = threadIdx.x;
  __shared__ unsigned long long red[1024];
  for (int p = 0; p < 2 * NI; ++p) {
    const bool bot = (p >= NI);
    float* buf = bot ? vb : vt;
    unsigned long long best = 0ull;
    for (int i = tid; i < Nn; i += 1024) {
      float v = buf[i];
      unsigned o = ordf(bot ? -v : v);
      unsigned long long key = ((unsigned long long)o << 32) | (unsigned)i;
      best = (key > best) ? key : best;
    }
    red[tid] = best;
    __syncthreads();
    for (int s = 512; s > 0; s >>= 1) {
      if (tid < s && red[tid + s] > red[tid]) red[tid] = red[tid + s];
      __syncthreads();
    }
    if (tid == 0) {
      int ii = (int)(red[0] & 0xFFFFFFFFull);
      idx[p] = ii;
      buf[ii] = bot ? INFINITY : -INFINITY;
      __threadfence_block();
    }
    __syncthreads();
  }
}

// ---------------- instance-level smooth top-1 SVM loss
__global__ __launch_bounds__(128) void k_inst(const float* __restrict__ h,
                                              const float* __restrict__ Winst,
                                              const float* __restrict__ binst,
                                              const int* __restrict__ plabel,
                                              const int* __restrict__ idx,
                                              float* __restrict__ out_loss) {
  const int t = threadIdx.x;
  const int lab = *plabel;
  float loss = 0.f;
  if (t < 2 * NI) {
    const int ii = idx[t];
    const float* hr = h + (size_t)ii * Ll;
    const float* w0 = Winst + (size_t)(lab * 2 + 0) * Ll;
    const float* w1 = Winst + (size_t)(lab * 2 + 1) * Ll;
    float l0 = binst[lab * 2 + 0], l1 = binst[lab * 2 + 1];
#pragma unroll 4
    for (int k = 0; k < Ll; ++k) {
      float hv = hr[k];
      l0 += hv * w0[k];
      l1 += hv * w1[k];
    }
    const int tgt = (t < NI) ? 1 : 0;
    float z0 = l0 + ((tgt == 0) ? 0.f : 1.f);
    float z1 = l1 + ((tgt == 1) ? 0.f : 1.f);
    float m = fmaxf(z0, z1);
    float lse = m + logf(expf(z0 - m) + expf(z1 - m));
    float sy = (tgt == 0) ? l0 : l1;
    loss = lse - sy;
  }
  __shared__ float s[128];
  s[t] = loss;
  __syncthreads();
  for (int st = 64; st > 0; st >>= 1) {
    if (t < st) s[t] += s[t + st];
    __syncthreads();
  }
  if (t == 0) out_loss[0] = s[0] / (float)(2 * NI);
}

// ---------------- bag logits partials
__global__ __launch_bounds__(256) void k_bag(const float* __restrict__ h,
                                             const float* __restrict__ A,
                                             const float* __restrict__ Wbag,
                                             const float* __restrict__ mx,
                                             const float* __restrict__ sm,
                                             float* __restrict__ plog) {
  __shared__ float wb[2 * Ll];
  for (int j = threadIdx.x; j < 2 * Ll; j += 256) wb[j] = Wbag[j];
  __syncthreads();
  const int i = blockIdx.x * 256 + threadIdx.x;  // exactly N threads total
  const float4* hr = (const float4*)(h + (size_t)i * Ll);
  const float4* w0 = (const float4*)(wb);
  const float4* w1 = (const float4*)(wb + Ll);
  float p0 = 0.f, p1 = 0.f;
#pragma unroll 4
  for (int k = 0; k < Ll / 4; ++k) {
    float4 v = hr[k];
    float4 x0 = w0[k], x1 = w1[k];
    p0 += v.x * x0.x + v.y * x0.y + v.z * x0.z + v.w * x0.w;
    p1 += v.x * x1.x + v.y * x1.y + v.z * x1.z + v.w * x1.w;
  }
  const float att0 = expf(A[2 * i] - mx[0]) / sm[0];
  const float att1 = expf(A[2 * i + 1] - mx[1]) / sm[1];
  float c0 = att0 * p0, c1 = att1 * p1;
  __shared__ float s0[256], s1[256];
  s0[threadIdx.x] = c0;
  s1[threadIdx.x] = c1;
  __syncthreads();
  for (int s = 128; s > 0; s >>= 1) {
    if (threadIdx.x < s) {
      s0[threadIdx.x] += s0[threadIdx.x + s];
      s1[threadIdx.x] += s1[threadIdx.x + s];
    }
    __syncthreads();
  }
  if (threadIdx.x == 0) {
    plog[blockIdx.x * 2] = s0[0];
    plog[blockIdx.x * 2 + 1] = s1[0];
  }
}

__global__ __launch_bounds__(256) void k_fin(const float* __restrict__ plog,
                                             const float* __restrict__ bbag,
                                             float* __restrict__ out) {
  const int t = threadIdx.x;
  __shared__ float s0[256], s1[256];
  s0[t] = plog[2 * t] + plog[2 * (t + 256)];
  s1[t] = plog[2 * t + 1] + plog[2 * (t + 256) + 1];
  __syncthreads();
  for (int s = 128; s > 0; s >>= 1) {
    if (t < s) {
      s0[t] += s0[t + s];
      s1[t] += s1[t + s];
    }
    __syncthreads();
  }
  if (t == 0) {
    float L0 = s0[0] + bbag[0];
    float L1 = s1[0] + bbag[1];
    out[0] = L0;
    out[1] = L1;
    float m = fmaxf(L0, L1);
    float e0 = expf(L0 - m), e1 = expf(L1 - m);
    float inv = 1.f / (e0 + e1);
    out[2] = e0 * inv;
    out[3] = e1 * inv;
  }
}

extern "C" void kernel_launch(void* const* d_in, const int* in_sizes, int n_in,
                              void* d_out, int out_size, void* d_ws,
                              size_t ws_size, hipStream_t stream) {
  const float* x = (const float*)d_in[0];
  const int* plabel = (const int*)d_in[1];
  const float* Wc = (const float*)d_in[2];
  const float* bc = (const float*)d_in[3];
  const float* Wv = (const float*)d_in[4];
  const float* bv = (const float*)d_in[5];
  const float* Wu = (const float*)d_in[6];
  const float* bu = (const float*)d_in[7];
  const float* Wa = (const float*)d_in[8];
  const float* ba = (const float*)d_in[9];
  const float* Winst = (const float*)d_in[10];
  const float* binst = (const float*)d_in[11];
  const float* Wbag = (const float*)d_in[12];
  const float* bbag = (const float*)d_in[13];

  float* ws = (float*)d_ws;
  const size_t OFF_H = 0;
  const size_t OFF_G = OFF_H + (size_t)Nn * Ll;
  const size_t OFF_A = OFF_G + (size_t)Nn * Dd;
  const size_t OFF_VT = OFF_A + (size_t)Nn * 2;
  const size_t OFF_VB = OFF_VT + (size_t)Nn;
  const size_t OFF_IDX = OFF_VB + (size_t)Nn;
  const size_t OFF_PMAX = OFF_IDX + 128;
  const size_t OFF_PSUM = OFF_PMAX + 512;
  const size_t OFF_MAX = OFF_PSUM + 512;
  const size_t OFF_SUM = OFF_MAX + 2;
  const size_t OFF_PLOG = OFF_SUM + 2;

  float* h = ws + OFF_H;
  float* g = ws + OFF_G;
  float* A = ws + OFF_A;
  float* vt = ws + OFF_VT;
  float* vb = ws + OFF_VB;
  int* idx = (int*)(ws + OFF_IDX);
  float* pmax = ws + OFF_PMAX;
  float* psum = ws + OFF_PSUM;
  float* mx = ws + OFF_MAX;
  float* sm = ws + OFF_SUM;
  float* plog = ws + OFF_PLOG;
  float* out = (float*)d_out;

  k_gemm_h<<<dim3(Ll / 64, Nn / 128, 1), 256, 0, stream>>>(x, Wc, bc, h);
  k_gemm_g<<<dim3(Dd / 64, Nn / 128, 1), 256, 0, stream>>>(h, Wv, bv, Wu, bu, g);
  k_proj<<<Nn / 256, 256, 0, stream>>>(g, Wa, ba, plabel, A, vt, vb, out + 4);
  k_max1<<<256, 256, 0, stream>>>(A, pmax);
  k_max2<<<1, 256, 0, stream>>>(pmax, mx);
  k_sum1<<<256, 256, 0, stream>>>(A, mx, psum);
  k_sum2<<<1, 256, 0, stream>>>(psum, sm);
  k_topk<<<1, 1024, 0, stream>>>(vt, vb, idx);
  k_inst<<<1, 128, 0, stream>>>(h, Winst, binst, plabel, idx,
                                out + 4 + 2 * (size_t)Nn);
  k_bag<<<Nn / 256, 256, 0, stream>>>(h, A, Wbag, mx, sm, plog);
  k_fin<<<1, 256, 0, stream>>>(plog, bbag, out);
}